// feedback_controller_56753697849632
// MI455X (gfx1250) — compile-verified
//
#include <hip/hip_runtime.h>
#include <math.h>

// ---- problem constants (from reference) ----
constexpr int T_STEPS = 4096;
constexpr int BATCH   = 9;
constexpr int NHID    = 500;
constexpr int NMUS    = 6;
constexpr int NHP     = 512;   // padded hidden
constexpr int HSTR    = 516;   // LDS row stride (floats) -> bank-conflict-free
constexpr int DELAYN  = 6;     // DELAY+1

typedef float v2f __attribute__((ext_vector_type(2)));
typedef float v4f __attribute__((ext_vector_type(4)));
typedef float v8f __attribute__((ext_vector_type(8)));

// ---------------------------------------------------------------------------
// Workspace layout (floats):
//   ws[0        .. 262143]  W_in_self  swizzled 32 tiles x 8192
//   ws[262144   .. 524287]  W_out      swizzled
//   ws[524288   .. 786431]  W_out_self swizzled
//   ws[786432   .. 794623]  W_in       padded row-major 512x16
//
// Swizzle: tile t (output cols n0=16t), K-pair p (k-steps 2p,2p+1), lane l:
//   float4 at [t*8192 + p*128 + l*4] =
//     { W[j][8p+2hi], W[j][8p+2hi+1], W[j][8p+4+2hi], W[j][8p+4+2hi+1] }
//   with j = 16t + (l%16), hi = l/16.
// One global_load_b128 per lane then feeds TWO v_wmma_f32_16x16x4_f32.
// ---------------------------------------------------------------------------
__global__ void pack_weights_k(const float* __restrict__ w_in,
                               const float* __restrict__ w_in_self,
                               const float* __restrict__ w_out,
                               const float* __restrict__ w_out_self,
                               float* __restrict__ ws)
{
    const int SQ = NHP * NHP;
    int idx = blockIdx.x * blockDim.x + threadIdx.x;
    if (idx < 3 * SQ) {
        int region = idx / SQ;
        int r = idx % SQ;
        int t = r >> 13;          // tile (0..31)
        int w = r & 8191;
        int p = w >> 7;           // K-pair (0..63)
        int u = w & 127;
        int l = u >> 2;           // lane (0..31)
        int q = u & 3;            // float within float4
        int j  = t * 16 + (l & 15);
        int hi = l >> 4;
        int kk = 8 * p + ((q >> 1) << 2) + 2 * hi + (q & 1);
        const float* src = (region == 0) ? w_in_self : (region == 1) ? w_out : w_out_self;
        ws[idx] = (j < NHID && kk < NHID) ? src[j * NHID + kk] : 0.0f;
    } else {
        int r = idx - 3 * SQ;
        if (r < NHP * 16) {
            int j = r >> 4, k = r & 15;
            ws[3 * SQ + r] = (j < NHID && k < 12) ? w_in[j * 12 + k] : 0.0f;
        }
    }
}

// ---------------------------------------------------------------------------
// Persistent single-workgroup recurrent kernel.
// 1024 threads = 32 wave32s; wave w owns output-neuron tile n0 = 16*w.
// Hidden state in LDS; weights streamed (L2-resident) via b128 loads.
// ---------------------------------------------------------------------------
__global__ __launch_bounds__(1024, 1) void rnn_persistent(
    const float* __restrict__ pert,    // (T-1,B,2)
    const float* __restrict__ nh1,     // (T-1,B,NH)
    const float* __restrict__ nh2,     // (T-1,B,NH)
    const float* __restrict__ nmus,    // (T-1,B,NM)
    const float* __restrict__ home,    // (B,2)
    const float* __restrict__ b_in,    // (NH)
    const float* __restrict__ b_out,   // (NH)
    const float* __restrict__ wmus,    // (NM,NH)
    const float* __restrict__ marm,    // (NM,2)
    const float* __restrict__ ws,      // packed weights
    float* __restrict__ out)           // (T,B,4)
{
    __shared__ __align__(16) float h1s[16 * HSTR];
    __shared__ __align__(16) float h2s[16 * HSTR];
    __shared__ __align__(16) float netin[16 * 20];
    __shared__ float mstate[2][16 * 8];
    __shared__ float bjb[DELAYN][16 * 4];
    __shared__ float bmb[DELAYN][16 * 8];
    __shared__ float jss[16 * 4];

    const int tid  = threadIdx.x;
    const int lane = tid & 31;
    const int wv   = tid >> 5;          // 0..31
    const int m16  = lane & 15;         // M (A) / N (B,D) sub-index
    const int hi   = lane >> 4;         // K-half selector
    const int n0   = wv * 16;           // this wave's output-column tile

    const float* Wis = ws;                      // W_in_self  (swizzled)
    const float* Wo  = ws + 262144;             // W_out      (swizzled)
    const float* Wos = ws + 2 * 262144;         // W_out_self (swizzled)
    const float* Wi  = ws + 3 * 262144;         // W_in       512x16 row-major

    const int tileoff = wv * 8192 + lane * 4;   // per-wave packed-tile base
    const int arow    = m16 * HSTR + 2 * hi;    // A-operand LDS row base

    // zero LDS (padded rows/cols of h must stay 0 forever)
    for (int i = tid; i < 16 * HSTR; i += 1024) { h1s[i] = 0.0f; h2s[i] = 0.0f; }
    for (int i = tid; i < 16 * 20; i += 1024) netin[i] = 0.0f;
    if (tid < 16 * 8) { mstate[0][tid] = 0.0f; mstate[1][tid] = 0.0f; }
    if (tid < 16 * 4) jss[tid] = 0.0f;
    for (int i = tid; i < DELAYN * 16 * 4; i += 1024) ((float*)bjb)[i] = 0.0f;
    for (int i = tid; i < DELAYN * 16 * 8; i += 1024) ((float*)bmb)[i] = 0.0f;
    __syncthreads();

    for (int s = 0; s < T_STEPS; ++s) {
        const bool init = (s == 0);
        const int ring  = s % DELAYN;     // delay slot to read (oldest) then overwrite
        const int mwr   = s & 1;          // m_state write slot
        const int mrd   = (s + 1) & 1;    // m_state read slot (previous step)

        // ---- phase 1: net_in = [0,0, 2*(jd_pos-home), 0.5*jd_vel, md(6), pad] ----
        if (!init && tid < 256) {
            int m = tid >> 4, k = tid & 15;
            float v = 0.0f;
            if (m < BATCH) {
                if (k >= 2 && k < 4)       v = 2.0f * (bjb[ring][m * 4 + (k - 2)] - home[m * 2 + (k - 2)]);
                else if (k >= 4 && k < 6)  v = 0.5f * bjb[ring][m * 4 + 2 + (k - 4)];
                else if (k >= 6 && k < 12) v = bmb[ring][m * 8 + (k - 6)];
            }
            netin[m * 20 + k] = v;
        }
        __syncthreads();

        // ---- phase 2: pre1 = net_in@W_in.T + h1p@W_in_self.T (WMMA) ----
        v8f acc;
        #pragma unroll
        for (int r = 0; r < 8; ++r) acc[r] = 0.0f;
        if (!init) {
            #pragma unroll
            for (int ks = 0; ks < 4; ++ks) {               // K = 16 (W_in)
                int kk = 4 * ks + 2 * hi;
                v2f a = *(const v2f*)&netin[m16 * 20 + kk];
                v2f b = *(const v2f*)&Wi[(n0 + m16) * 16 + kk];
                acc = __builtin_amdgcn_wmma_f32_16x16x4_f32(false, a, false, b,
                                                            (short)0, acc, false, false);
            }
            const float* wtile = Wis + tileoff;
            #pragma unroll 2
            for (int p = 0; p < 64; ++p) {                 // K = 512 (self)
                v4f wb = *(const v4f*)&wtile[p * 128];
                v2f a0 = *(const v2f*)&h1s[arow + 8 * p];
                v2f a1 = *(const v2f*)&h1s[arow + 8 * p + 4];
                v2f b0 = __builtin_shufflevector(wb, wb, 0, 1);
                v2f b1 = __builtin_shufflevector(wb, wb, 2, 3);
                acc = __builtin_amdgcn_wmma_f32_16x16x4_f32(false, a0, false, b0,
                                                            (short)0, acc, false, false);
                acc = __builtin_amdgcn_wmma_f32_16x16x4_f32(false, a1, false, b1,
                                                            (short)0, acc, false, false);
            }
        }
        __syncthreads();   // all waves done reading h1s

        // ---- phase 3: h1 = 0.5*tanh(pre1+b_in) + 0.5*h1p + n1*h1p^2 (in place) ----
        {
            int n = n0 + m16;
            if (n < NHID) {
                float bi = b_in[n];
                #pragma unroll
                for (int r = 0; r < 8; ++r) {
                    int m = r + 8 * hi;                     // D layout: M = r + 8*(lane/16)
                    if (m < BATCH) {
                        float th = tanhf(acc[r] + bi);
                        float hp = h1s[m * HSTR + n];
                        float hv;
                        if (init) hv = th;
                        else {
                            float nn = nh1[((size_t)(s - 1) * BATCH + m) * NHID + n];
                            hv = 0.5f * th + 0.5f * hp + nn * hp * hp;
                        }
                        h1s[m * HSTR + n] = hv;
                    }
                }
            }
        }
        __syncthreads();

        // ---- phase 4: pre2 = h1@W_out.T + h2p@W_out_self.T (WMMA) ----
        #pragma unroll
        for (int r = 0; r < 8; ++r) acc[r] = 0.0f;
        {
            const float* wtile = Wo + tileoff;
            #pragma unroll 2
            for (int p = 0; p < 64; ++p) {
                v4f wb = *(const v4f*)&wtile[p * 128];
                v2f a0 = *(const v2f*)&h1s[arow + 8 * p];
                v2f a1 = *(const v2f*)&h1s[arow + 8 * p + 4];
                v2f b0 = __builtin_shufflevector(wb, wb, 0, 1);
                v2f b1 = __builtin_shufflevector(wb, wb, 2, 3);
                acc = __builtin_amdgcn_wmma_f32_16x16x4_f32(false, a0, false, b0,
                                                            (short)0, acc, false, false);
                acc = __builtin_amdgcn_wmma_f32_16x16x4_f32(false, a1, false, b1,
                                                            (short)0, acc, false, false);
            }
        }
        if (!init) {
            const float* wtile = Wos + tileoff;
            #pragma unroll 2
            for (int p = 0; p < 64; ++p) {
                v4f wb = *(const v4f*)&wtile[p * 128];
                v2f a0 = *(const v2f*)&h2s[arow + 8 * p];
                v2f a1 = *(const v2f*)&h2s[arow + 8 * p + 4];
                v2f b0 = __builtin_shufflevector(wb, wb, 0, 1);
                v2f b1 = __builtin_shufflevector(wb, wb, 2, 3);
                acc = __builtin_amdgcn_wmma_f32_16x16x4_f32(false, a0, false, b0,
                                                            (short)0, acc, false, false);
                acc = __builtin_amdgcn_wmma_f32_16x16x4_f32(false, a1, false, b1,
                                                            (short)0, acc, false, false);
            }
        }
        __syncthreads();

        // ---- phase 5: h2 update (in place) ----
        {
            int n = n0 + m16;
            if (n < NHID) {
                float bo = b_out[n];
                #pragma unroll
                for (int r = 0; r < 8; ++r) {
                    int m = r + 8 * hi;
                    if (m < BATCH) {
                        float th = tanhf(acc[r] + bo);
                        float hp = h2s[m * HSTR + n];
                        float hv;
                        if (init) hv = th;
                        else {
                            float nn = nh2[((size_t)(s - 1) * BATCH + m) * NHID + n];
                            hv = 0.5f * th + 0.5f * hp + nn * hp * hp;
                        }
                        h2s[m * HSTR + n] = hv;
                    }
                }
            }
        }
        __syncthreads();

        // ---- phase 6: m = 0.2*relu(h2@W_mus.T) + 0.8*mp + nm*mp^2 ----
        if (tid < BATCH * NMUS) {
            int mb = tid / NMUS, j = tid % NMUS;
            const float* wr = &wmus[j * NHID];
            const float* hr = &h2s[mb * HSTR];
            float dot = 0.0f;
            #pragma unroll 4
            for (int k = 0; k < NHID; ++k) dot += hr[k] * wr[k];
            float rl = dot > 0.0f ? dot : 0.0f;
            float mv;
            if (init) mv = rl;
            else {
                float mp = mstate[mrd][mb * 8 + j];
                float nn = nmus[((size_t)(s - 1) * BATCH + mb) * NMUS + j];
                mv = 0.2f * rl + 0.8f * mp + nn * mp * mp;
            }
            mstate[mwr][mb * 8 + j] = mv;
            if (init) {
                for (int q = 0; q < DELAYN; ++q) bmb[q][mb * 8 + j] = mv;
            } else {
                bmb[ring][mb * 8 + j] = mv;
            }
        }
        __syncthreads();

        // ---- phase 7: torque -> joint integration -> output ----
        if (tid < BATCH * 2) {
            int b = tid >> 1, d = tid & 1;
            if (init) {
                float p = home[b * 2 + d];
                jss[b * 4 + d] = p; jss[b * 4 + 2 + d] = 0.0f;
                for (int q = 0; q < DELAYN; ++q) { bjb[q][b * 4 + d] = p; bjb[q][b * 4 + 2 + d] = 0.0f; }
                out[b * 4 + d] = p;
                out[b * 4 + 2 + d] = 0.0f;
            } else {
                float tq = pert[((size_t)(s - 1) * BATCH + b) * 2 + d];
                #pragma unroll
                for (int j = 0; j < NMUS; ++j) tq += mstate[mwr][b * 8 + j] * marm[j * 2 + d];
                float vel = jss[b * 4 + 2 + d] + 0.01f * tq;
                float pos = jss[b * 4 + d] + 0.01f * vel;
                jss[b * 4 + d] = pos; jss[b * 4 + 2 + d] = vel;
                bjb[ring][b * 4 + d] = pos; bjb[ring][b * 4 + 2 + d] = vel;
                out[(size_t)s * BATCH * 4 + b * 4 + d] = pos;
                out[(size_t)s * BATCH * 4 + b * 4 + 2 + d] = vel;
            }
        }
        __syncthreads();
    }
}

extern "C" void kernel_launch(void* const* d_in, const int* in_sizes, int n_in,
                              void* d_out, int out_size, void* d_ws, size_t ws_size,
                              hipStream_t stream) {
    (void)in_sizes; (void)n_in; (void)out_size; (void)ws_size;
    // inputs in setup_inputs() order; d_in[0] (des_targ) is multiplied by 0.0
    // in the reference -> never used.
    const float* pert   = (const float*)d_in[1];
    const float* n_h1   = (const float*)d_in[2];
    const float* n_h2   = (const float*)d_in[3];
    const float* n_m    = (const float*)d_in[4];
    const float* home   = (const float*)d_in[5];
    const float* w_in   = (const float*)d_in[6];
    const float* b_in   = (const float*)d_in[7];
    const float* w_insf = (const float*)d_in[8];
    const float* w_out  = (const float*)d_in[9];
    const float* b_out  = (const float*)d_in[10];
    const float* w_outsf= (const float*)d_in[11];
    const float* w_mus  = (const float*)d_in[12];
    const float* m_arm  = (const float*)d_in[13];
    float* out = (float*)d_out;
    float* ws  = (float*)d_ws;

    const int total = 3 * NHP * NHP + NHP * 16;   // 794624 floats (~3.03 MB)
    pack_weights_k<<<(total + 1023) / 1024, 1024, 0, stream>>>(w_in, w_insf, w_out, w_outsf, ws);
    rnn_persistent<<<1, 1024, 0, stream>>>(pert, n_h1, n_h2, n_m, home,
                                           b_in, b_out, w_mus, m_arm, ws, out);
}